// DeformableCaps2d_70909910057282
// MI455X (gfx1250) — compile-verified
//
#include <hip/hip_runtime.h>
#include <hip/hip_bf16.h>
#include <math.h>

typedef __attribute__((ext_vector_type(2))) float v2f;
typedef __attribute__((ext_vector_type(8))) float v8f;

#define NUM_SAMPLES 4
#define NUM_CHILD   8
#define DIM_CHILD   8
#define NUM_PARENT  8
#define DIM_PARENT  16
#define Hh 32
#define Ww 32
#define Cc 256
#define Bb 8
#define Mtot (Bb*Hh*Ww)          // 8192
#define CAPS_CH (NUM_CHILD*DIM_CHILD)  // 64

// ---------------------------------------------------------------------------
// Kernel 1: PrimaryCaps 1x1 conv as WMMA f32 GEMM [8192x256]x[256x64],
// fused BN(inference) + SiLU.  Output P[m][64] row-major.
// grid.x = 512 m-tiles; block = 128 threads = 4 waves, wave w -> n-tile w*16.
// ---------------------------------------------------------------------------
__global__ __launch_bounds__(128)
void primary_gemm_kernel(const float* __restrict__ x,      // [B,256,32,32]
                         const float* __restrict__ w,      // [64,256]
                         const float* __restrict__ gamma,
                         const float* __restrict__ beta,
                         const float* __restrict__ mean,
                         const float* __restrict__ var,
                         float* __restrict__ P)             // [8192,64]
{
    const int mtile = blockIdx.x * 16;
    const int wave  = threadIdx.x >> 5;
    const int lane  = threadIdx.x & 31;
    const int ntile = wave * 16;
    const int half  = lane >> 4;      // 0: K pair (0,1), 1: K pair (2,3)
    const int lm    = lane & 15;

    const int m  = mtile + lm;
    const int b  = m >> 10;
    const int sp = m & 1023;
    // A[m,k] = x[(b*256 + k)*1024 + sp]
    const float* __restrict__ xa = x + (size_t)b * Cc * 1024 + sp;
    // B[k,n] = w[n*256 + k]
    const float* __restrict__ wb = w + (size_t)(ntile + lm) * Cc;

    v8f acc = {0.f,0.f,0.f,0.f,0.f,0.f,0.f,0.f};
    for (int k0 = 0; k0 < Cc; k0 += 4) {
        const int k = k0 + 2 * half;
        v2f afrag, bfrag;
        afrag.x = xa[(size_t)k * 1024];
        afrag.y = xa[(size_t)(k + 1) * 1024];
        bfrag.x = wb[k];
        bfrag.y = wb[k + 1];
        acc = __builtin_amdgcn_wmma_f32_16x16x4_f32(
            /*neg_a=*/false, afrag, /*neg_b=*/false, bfrag,
            /*c_mod=*/(short)0, acc, /*reuse_a=*/false, /*reuse_b=*/false);
    }

    const int o = ntile + lm;        // output channel (fixed per lane)
    const float invstd = rsqrtf(var[o] + 1e-5f);
    const float g  = gamma[o];
    const float bt = beta[o];
    const float mu = mean[o];
#pragma unroll
    for (int v = 0; v < 8; ++v) {
        const int mr = mtile + v + 8 * half;
        float val = (acc[v] - mu) * invstd * g + bt;
        val = val / (1.f + __expf(-val));          // SiLU
        P[(size_t)mr * CAPS_CH + o] = val;
    }
}

// ---------------------------------------------------------------------------
// Kernel 2: squash over DIM_CHILD=8 within each capsule, in place on P.
// thread per (m, capsule): 65536 threads.
// ---------------------------------------------------------------------------
__global__ __launch_bounds__(256)
void squash_kernel(float* __restrict__ P)
{
    const int gid = blockIdx.x * blockDim.x + threadIdx.x;
    if (gid >= Mtot * NUM_CHILD) return;
    const int m   = gid >> 3;
    const int cap = gid & 7;
    float* p = P + (size_t)m * CAPS_CH + cap * DIM_CHILD;
    float v[8];
    float sq = 0.f;
#pragma unroll
    for (int i = 0; i < 8; ++i) { v[i] = p[i]; sq += v[i] * v[i]; }
    const float scale = (sq / (1.f + sq)) * rsqrtf(sq + 1e-7f);
#pragma unroll
    for (int i = 0; i < 8; ++i) p[i] = v[i] * scale;
}

// ---------------------------------------------------------------------------
// Kernel 3: offset 3x3 conv (SAME, zero pad) + tanh + grid construction.
// Block per pixel (8192 blocks), 256 threads = one input channel each,
// LDS tree reduction over channels for the 8 outputs.
// grids layout: [B, S, H, W, 2]  (x,y)
// ---------------------------------------------------------------------------
__global__ __launch_bounds__(256)
void offset_grid_kernel(const float* __restrict__ x,       // [B,256,32,32]
                        const float* __restrict__ ow,      // [8,256,3,3]
                        const float* __restrict__ ob,      // [8]
                        float* __restrict__ grids)          // [B,4,1024,2]
{
    __shared__ float red[8 * 256];
    const int pix  = blockIdx.x;       // b*1024 + h*32 + w
    const int b    = pix >> 10;
    const int spos = pix & 1023;
    const int h    = spos >> 5;
    const int w    = spos & 31;
    const int c    = threadIdx.x;      // input channel

    float acc[8];
#pragma unroll
    for (int o = 0; o < 8; ++o) acc[o] = 0.f;

    const float* __restrict__ xc = x + ((size_t)b * Cc + c) * 1024;
#pragma unroll
    for (int dy = 0; dy < 3; ++dy) {
        const int yy = h + dy - 1;
        if (yy < 0 || yy > 31) continue;
#pragma unroll
        for (int dx = 0; dx < 3; ++dx) {
            const int xx = w + dx - 1;
            if (xx < 0 || xx > 31) continue;
            const float xv = xc[yy * 32 + xx];
#pragma unroll
            for (int o = 0; o < 8; ++o)
                acc[o] += xv * ow[(((size_t)o * Cc + c) * 3 + dy) * 3 + dx];
        }
    }
#pragma unroll
    for (int o = 0; o < 8; ++o) red[o * 256 + c] = acc[o];
    __syncthreads();
    for (int str = 128; str > 0; str >>= 1) {
        if (c < str) {
#pragma unroll
            for (int o = 0; o < 8; ++o)
                red[o * 256 + c] += red[o * 256 + c + str];
        }
        __syncthreads();
    }
    if (c < 8) {
        const int o = c;
        const int j = o & 1;           // 0 -> x, 1 -> y
        const int s = o >> 1;
        const float r   = red[o * 256] + ob[o];
        const float off = tanhf(r) * (2.f / 31.f);   // /((W-1)/2), OFFSET_SCALE=1
        const float base = (j == 0) ? (-1.f + 2.f * (float)w / 31.f)
                                    : (-1.f + 2.f * (float)h / 31.f);
        grids[((size_t)(b * NUM_SAMPLES + s) * 1024 + spos) * 2 + j] = base + off;
    }
}

// ---------------------------------------------------------------------------
// Kernel 4: bilinear sampling (align_corners=True, zeros padding) of the
// 64-channel child field at the 4 deformed grids.
// thread per (m, s): 32768 threads.  sampled layout: [m][s*64 + ch]
// ---------------------------------------------------------------------------
__global__ __launch_bounds__(256)
void sample_kernel(const float* __restrict__ child,    // [8192,64]
                   const float* __restrict__ grids,    // [B,4,1024,2]
                   float* __restrict__ sampled)         // [8192,256]
{
    const int gid = blockIdx.x * blockDim.x + threadIdx.x;
    if (gid >= Mtot * NUM_SAMPLES) return;
    const int s = gid & 3;
    const int m = gid >> 2;
    const int b    = m >> 10;
    const int spos = m & 1023;

    const float gx = grids[((size_t)(b * NUM_SAMPLES + s) * 1024 + spos) * 2 + 0];
    const float gy = grids[((size_t)(b * NUM_SAMPLES + s) * 1024 + spos) * 2 + 1];
    const float px = (gx + 1.f) * 0.5f * 31.f;
    const float py = (gy + 1.f) * 0.5f * 31.f;
    const float x0 = floorf(px), y0 = floorf(py);
    const float wx1 = px - x0, wx0 = 1.f - wx1;
    const float wy1 = py - y0, wy0 = 1.f - wy1;

    float cw[4];
    int   cm[4];
#pragma unroll
    for (int cor = 0; cor < 4; ++cor) {
        const float xf = x0 + (float)(cor & 1);
        const float yf = y0 + (float)(cor >> 1);
        const bool valid = (xf >= 0.f) && (xf <= 31.f) && (yf >= 0.f) && (yf <= 31.f);
        int xi = (int)fminf(fmaxf(xf, 0.f), 31.f);
        int yi = (int)fminf(fmaxf(yf, 0.f), 31.f);
        const float wgt = ((cor & 1) ? wx1 : wx0) * ((cor >> 1) ? wy1 : wy0);
        cw[cor] = valid ? wgt : 0.f;
        cm[cor] = b * 1024 + yi * 32 + xi;
    }

    const float4* __restrict__ c4 = (const float4*)child;
    float4* __restrict__ out4 = (float4*)(sampled + (size_t)m * 256 + s * 64);
#pragma unroll
    for (int ch = 0; ch < CAPS_CH / 4; ++ch) {
        float4 a = c4[(size_t)cm[0] * 16 + ch];
        float4 bq = c4[(size_t)cm[1] * 16 + ch];
        float4 cq = c4[(size_t)cm[2] * 16 + ch];
        float4 dq = c4[(size_t)cm[3] * 16 + ch];
        float4 r;
        r.x = cw[0]*a.x + cw[1]*bq.x + cw[2]*cq.x + cw[3]*dq.x;
        r.y = cw[0]*a.y + cw[1]*bq.y + cw[2]*cq.y + cw[3]*dq.y;
        r.z = cw[0]*a.z + cw[1]*bq.z + cw[2]*cq.z + cw[3]*dq.z;
        r.w = cw[0]*a.w + cw[1]*bq.w + cw[2]*cq.w + cw[3]*dq.w;
        out4[ch] = r;
    }
}

// ---------------------------------------------------------------------------
// Kernel 5: fused u_hat + 3-iteration dynamic routing, one block per m.
// u_hat (32n x 8o x 16d = 4096 f32 = 16KB) lives in LDS; output [B,128,H,W].
// ---------------------------------------------------------------------------
__global__ __launch_bounds__(256)
void routing_kernel(const float* __restrict__ sampled,   // [8192,256]
                    const float* __restrict__ rW,        // [32,8,16,8]
                    float* __restrict__ out)              // [8,128,1024]
{
    __shared__ float sin_s[256];        // sampled row
    __shared__ float u_s[32 * 128];     // u_hat[n][o*16+d]
    __shared__ float blog[32 * 8];      // routing logits
    __shared__ float c_s[32 * 8];       // coupling coeffs
    __shared__ float su[128];           // s[o*16+d]
    __shared__ float vsm[128];          // v[o*16+d]
    __shared__ float vscale[8];

    const int m = blockIdx.x;
    const int t = threadIdx.x;

    sin_s[t] = sampled[(size_t)m * 256 + t];
    if (t < 256) blog[t & 255] = 0.f;   // all 256 logits
    __syncthreads();

    // u_hat: 4096 dot-8s, 16 per thread
#pragma unroll
    for (int j = 0; j < 16; ++j) {
        const int idx = j * 256 + t;          // n*128 + od
        const int n  = idx >> 7;
        const int od = idx & 127;
        const float* __restrict__ wrow = rW + (size_t)(n * 128 + od) * 8;
        const float* __restrict__ sv   = sin_s + n * 8;
        float u = 0.f;
#pragma unroll
        for (int i = 0; i < 8; ++i) u += wrow[i] * sv[i];
        u_s[idx] = u;
    }
    __syncthreads();

    for (int it = 0; it < 3; ++it) {
        // softmax over the 8 parents (one thread per n)
        if (t < 32) {
            float mx = blog[t * 8];
#pragma unroll
            for (int o = 1; o < 8; ++o) mx = fmaxf(mx, blog[t * 8 + o]);
            float sum = 0.f;
            float e[8];
#pragma unroll
            for (int o = 0; o < 8; ++o) { e[o] = __expf(blog[t * 8 + o] - mx); sum += e[o]; }
            const float inv = 1.f / sum;
#pragma unroll
            for (int o = 0; o < 8; ++o) c_s[t * 8 + o] = e[o] * inv;
        }
        __syncthreads();

        // s[o][d] = sum_n c[n][o] * u_hat[n][o*16+d]
        if (t < 128) {
            const int o = t >> 4;
            float acc = 0.f;
#pragma unroll
            for (int n = 0; n < 32; ++n)
                acc += c_s[n * 8 + o] * u_s[n * 128 + t];
            su[t] = acc;
        }
        __syncthreads();

        // squash per parent
        if (t < 8) {
            float sq = 0.f;
#pragma unroll
            for (int d = 0; d < 16; ++d) { const float v = su[t * 16 + d]; sq += v * v; }
            vscale[t] = (sq / (1.f + sq)) * rsqrtf(sq + 1e-7f);
        }
        __syncthreads();
        if (t < 128) vsm[t] = su[t] * vscale[t >> 4];
        __syncthreads();

        if (it < 2) {
            // b[n][o] += <u_hat[n][o], v[o]>
            const int n = t >> 3;
            const int o = t & 7;
            float dot = 0.f;
#pragma unroll
            for (int d = 0; d < 16; ++d)
                dot += u_s[n * 128 + o * 16 + d] * vsm[o * 16 + d];
            blog[t] += dot;
            __syncthreads();
        }
    }

    if (t < 128) {
        const int b    = m >> 10;
        const int spos = m & 1023;
        out[((size_t)b * 128 + t) * 1024 + spos] = vsm[t];
    }
}

// ---------------------------------------------------------------------------
extern "C" void kernel_launch(void* const* d_in, const int* in_sizes, int n_in,
                              void* d_out, int out_size, void* d_ws, size_t ws_size,
                              hipStream_t stream) {
    const float* x        = (const float*)d_in[0];
    const float* primaryW = (const float*)d_in[1];
    const float* bn_gamma = (const float*)d_in[2];
    const float* bn_beta  = (const float*)d_in[3];
    const float* bn_mean  = (const float*)d_in[4];
    const float* bn_var   = (const float*)d_in[5];
    const float* offset_w = (const float*)d_in[6];
    const float* offset_b = (const float*)d_in[7];
    const float* routingW = (const float*)d_in[8];
    float* out = (float*)d_out;

    float* wsf     = (float*)d_ws;
    float* P       = wsf;                       // 8192*64   = 524288 f
    float* grids   = wsf + 524288;              // 8*4*1024*2 = 65536 f
    float* sampled = wsf + 524288 + 65536;      // 8192*256  = 2097152 f

    // 1) PrimaryCaps GEMM + BN + SiLU  (WMMA f32 16x16x4)
    primary_gemm_kernel<<<Mtot / 16, 128, 0, stream>>>(
        x, primaryW, bn_gamma, bn_beta, bn_mean, bn_var, P);

    // 2) squash per capsule (in place -> child field)
    squash_kernel<<<(Mtot * NUM_CHILD) / 256, 256, 0, stream>>>(P);

    // 3) offset conv + grid construction
    offset_grid_kernel<<<Mtot, 256, 0, stream>>>(x, offset_w, offset_b, grids);

    // 4) deformable bilinear sampling
    sample_kernel<<<(Mtot * NUM_SAMPLES) / 256, 256, 0, stream>>>(P, grids, sampled);

    // 5) fused u_hat + dynamic routing
    routing_kernel<<<Mtot, 256, 0, stream>>>(sampled, routingW, out);
}